// MyGCN_10488310137582
// MI455X (gfx1250) — compile-verified
//
#include <hip/hip_runtime.h>

#define D        128
#define DOUT     8
#define NGRAPH   512
#define BN_EPS   1e-5f
#define LDS_PITCH 132   // row pitch (words): 16B-aligned rows for async B128, stride%64=4 -> conflict-free frags
#define MT       5      // M tiles (of 16 rows) per block
#define ROWS     (MT * 16)

typedef __attribute__((ext_vector_type(2))) float v2f;
typedef __attribute__((ext_vector_type(8))) float v8f;
typedef int v4i_ __attribute__((vector_size(4 * sizeof(int))));
typedef __attribute__((address_space(1))) v4i_* gptr_v4i;
typedef __attribute__((address_space(3))) v4i_* lptr_v4i;

#if __has_builtin(__builtin_amdgcn_global_load_async_to_lds_b128) && \
    __has_builtin(__builtin_amdgcn_s_wait_asynccnt)
#define USE_ASYNC_LDS 1
#else
#define USE_ASYNC_LDS 0
#endif

// ---------------------------------------------------------------- utilities
__global__ void k_fill(float* __restrict__ p, float v, int n) {
    int i = blockIdx.x * blockDim.x + threadIdx.x;
    if (i < n) p[i] = v;
}

// weighted in-degree: deg[col] += ew  (deg pre-filled with 1.0 = self-loop)
__global__ void k_deg(const int* __restrict__ col, const float* __restrict__ ew,
                      float* __restrict__ deg, int E) {
    int e = blockIdx.x * blockDim.x + threadIdx.x;
    if (e < E) atomicAdd(&deg[col[e]], ew[e]);
}

__global__ void k_dinv(float* __restrict__ deg, int n) {
    int i = blockIdx.x * blockDim.x + threadIdx.x;
    if (i < n) {
        float d = deg[i];
        deg[i] = d > 0.f ? rsqrtf(d) : 0.f;
    }
}

// ------------------------------------------------- WMMA f32 GEMM: C = A @ W
// A: [M,128] row-major, W: [128,128] row-major, C: [M,128].
// Block = 8 waves; block covers 80 rows (5 M-tiles); wave w owns N-tile w.
// A tile staged to LDS via async global->LDS DMA (no VGPR round-trip);
// each wave keeps 5 accumulators and reuses every B fragment 5x.
__global__ __launch_bounds__(256)
void k_gemm128(const float* __restrict__ A, const float* __restrict__ W,
               float* __restrict__ C, int M) {
    __shared__ float As[ROWS * LDS_PITCH];
    const int m0 = blockIdx.x * ROWS;

    // stage 80x128 A tile into LDS (16 bytes per lane per iteration)
    for (int t = threadIdx.x; t < ROWS * 32; t += 256) {
        int r  = t >> 5;
        int kc = (t & 31) << 2;
        int gr = m0 + r;
        if (gr >= M) gr = M - 1;                       // clamp (dup row, stores guarded)
#if USE_ASYNC_LDS
        __builtin_amdgcn_global_load_async_to_lds_b128(
            (gptr_v4i)(A + (size_t)gr * D + kc),
            (lptr_v4i)&As[r * LDS_PITCH + kc],
            0, 0);
#else
        float4 v = ((const float4*)(A + (size_t)gr * D))[t & 31];
        float* dst = &As[r * LDS_PITCH + kc];
        dst[0] = v.x; dst[1] = v.y; dst[2] = v.z; dst[3] = v.w;
#endif
    }
#if USE_ASYNC_LDS
    __builtin_amdgcn_s_wait_asynccnt(0);   // this wave's DMAs done; barrier publishes all
#endif
    __syncthreads();

    const int lane = threadIdx.x & 31;
    const int wv   = threadIdx.x >> 5;   // 0..7 -> N tile
    const int n0   = wv * 16;
    const int ml   = lane & 15;          // M row (A frag) / N col (B frag)
    const int kh   = lane >> 4;          // lanes 0-15: K 0,1 ; lanes 16-31: K 2,3
    const int colW = n0 + ml;

    v8f acc[MT];
    #pragma unroll
    for (int m = 0; m < MT; ++m) acc[m] = (v8f){};

    for (int kk0 = 0; kk0 < D; kk0 += 32) {           // 4 outer iterations
        // preload 8 k-steps of B fragments into registers (reused by 5 M-tiles)
        v2f bfrag[8];
        #pragma unroll
        for (int u = 0; u < 8; ++u) {
            int kb = kk0 + (u << 2) + 2 * kh;
            bfrag[u].x = W[kb * D + colW];
            bfrag[u].y = W[(kb + 1) * D + colW];
        }
        #pragma unroll
        for (int u = 0; u < 8; ++u) {
            int kb = kk0 + (u << 2) + 2 * kh;
            #pragma unroll
            for (int m = 0; m < MT; ++m) {
                const float* ap = &As[(m * 16 + ml) * LDS_PITCH + kb];
                v2f a;
                a.x = ap[0];
                a.y = ap[1];
                acc[m] = __builtin_amdgcn_wmma_f32_16x16x4_f32(
                    /*neg_a=*/false, a, /*neg_b=*/false, bfrag[u],
                    /*c_mod=*/(short)0, acc[m], /*reuse_a=*/false, /*reuse_b=*/false);
            }
        }
    }

    // C/D layout: lanes 0-15 -> rows j, lanes 16-31 -> rows j+8
    #pragma unroll
    for (int m = 0; m < MT; ++m) {
        int row0 = m0 + m * 16 + 8 * kh;
        #pragma unroll
        for (int j = 0; j < 8; ++j) {
            int row = row0 + j;
            if (row < M) C[(size_t)row * D + colW] = acc[m][j];
        }
    }
}

// ------------------------------------- edge aggregation: out[c] += norm * H[r]
// one wave per edge; each lane moves a float4 (32 lanes * 4 = 128 features)
__global__ void k_agg(const int* __restrict__ ei, const float* __restrict__ ew,
                      const float* __restrict__ dinv, const float* __restrict__ H,
                      float* __restrict__ out, int E) {
    int wid  = (blockIdx.x * blockDim.x + threadIdx.x) >> 5;
    int lane = threadIdx.x & 31;
    if (wid >= E) return;
    int r = ei[wid];
    int c = ei[(size_t)E + wid];
    float nw = dinv[r] * ew[wid] * dinv[c];
    float4 v = ((const float4*)(H + (size_t)r * D))[lane];
    float* orow = out + (size_t)c * D + lane * 4;
    atomicAdd(orow + 0, nw * v.x);
    atomicAdd(orow + 1, nw * v.y);
    atomicAdd(orow + 2, nw * v.z);
    atomicAdd(orow + 3, nw * v.w);
}

// ------------------- conv1 epilogue: self-loop + bias + BN(inference) + ReLU
__global__ void k_post1(float* __restrict__ agg, const float* __restrict__ h1,
                        const float* __restrict__ dinv, const float* __restrict__ b1,
                        const float* __restrict__ gamma, const float* __restrict__ beta,
                        const float* __restrict__ mu, const float* __restrict__ var,
                        int total) {
    int t = blockIdx.x * blockDim.x + threadIdx.x;
    if (t >= total) return;
    int i = t >> 7, d = t & (D - 1);
    float di  = dinv[i];
    float val = agg[t] + di * di * h1[t] + b1[d];
    val = gamma[d] * (val - mu[d]) * rsqrtf(var[d] + BN_EPS) + beta[d];
    agg[t] = fmaxf(val, 0.f);
}

// ---- conv2 epilogue: self-loop + bias, fused with global_add_pool (atomics)
__global__ void k_post2_pool(const float* __restrict__ agg, const float* __restrict__ g,
                             const float* __restrict__ dinv, const float* __restrict__ b2,
                             const int* __restrict__ batch, float* __restrict__ pooled,
                             int total) {
    int t = blockIdx.x * blockDim.x + threadIdx.x;
    if (t >= total) return;
    int i = t >> 7, d = t & (D - 1);
    float di  = dinv[i];
    float val = agg[t] + di * di * g[t] + b2[d];
    atomicAdd(&pooled[(size_t)batch[i] * D + d], val);
}

// ------------------------------------ final linear: out = pooled @ lin_W + b
__global__ void k_final(const float* __restrict__ pooled, const float* __restrict__ lw,
                        const float* __restrict__ lb, float* __restrict__ out) {
    int t = blockIdx.x * blockDim.x + threadIdx.x;
    if (t >= NGRAPH * DOUT) return;
    int gi = t >> 3, o = t & (DOUT - 1);
    float s = lb[o];
    #pragma unroll 8
    for (int d = 0; d < D; ++d) s += pooled[(size_t)gi * D + d] * lw[d * DOUT + o];
    out[t] = s;
}

extern "C" void kernel_launch(void* const* d_in, const int* in_sizes, int n_in,
                              void* d_out, int out_size, void* d_ws, size_t ws_size,
                              hipStream_t stream) {
    const float* x     = (const float*)d_in[0];
    const int*   ei    = (const int*)  d_in[1];   // [2, E]
    const float* ew    = (const float*)d_in[2];
    const int*   batch = (const int*)  d_in[3];
    const float* W1    = (const float*)d_in[4];
    const float* b1    = (const float*)d_in[5];
    const float* gam   = (const float*)d_in[6];
    const float* bet   = (const float*)d_in[7];
    const float* mu    = (const float*)d_in[8];
    const float* var   = (const float*)d_in[9];
    const float* W2    = (const float*)d_in[10];
    const float* b2    = (const float*)d_in[11];
    const float* lw    = (const float*)d_in[12];
    const float* lb    = (const float*)d_in[13];
    float* out = (float*)d_out;

    const int Nn = in_sizes[0] / D;   // 100000
    const int E  = in_sizes[1] / 2;   // 1600000
    const int ND = Nn * D;            // 12.8M

    // workspace partition (~103 MB)
    float* dinv   = (float*)d_ws;                 // Nn (rounded up)
    float* pooled = dinv + 102400;                // NGRAPH*D
    float* bufA   = pooled + NGRAPH * D;          // Nn*D
    float* bufB   = bufA + (size_t)ND;            // Nn*D

    const int TB = 256;
    dim3 blk(TB);
    const int gemmGrid = (Nn + ROWS - 1) / ROWS;  // 1250

    // 1) deg = 1 (self-loop weight), accumulate edge weights, rsqrt
    k_fill<<<(Nn + TB - 1) / TB, blk, 0, stream>>>(dinv, 1.0f, Nn);
    k_fill<<<(NGRAPH * D + TB - 1) / TB, blk, 0, stream>>>(pooled, 0.0f, NGRAPH * D);
    k_fill<<<(ND + TB - 1) / TB, blk, 0, stream>>>(bufB, 0.0f, ND);
    k_deg<<<(E + TB - 1) / TB, blk, 0, stream>>>(ei + (size_t)E, ew, dinv, E);
    k_dinv<<<(Nn + TB - 1) / TB, blk, 0, stream>>>(dinv, Nn);

    // 2) conv1: h1 = x @ W1 (WMMA), aggregate, BN+ReLU
    k_gemm128<<<gemmGrid, blk, 0, stream>>>(x, W1, bufA, Nn);
    {
        long long waves = (long long)E * 32;
        k_agg<<<(unsigned)((waves + TB - 1) / TB), blk, 0, stream>>>(ei, ew, dinv, bufA, bufB, E);
    }
    k_post1<<<(ND + TB - 1) / TB, blk, 0, stream>>>(bufB, bufA, dinv, b1, gam, bet, mu, var, ND);

    // 3) conv2: g = h2 @ W2 (WMMA), aggregate, fused epilogue + pooling
    k_gemm128<<<gemmGrid, blk, 0, stream>>>(bufB, W2, bufA, Nn);
    k_fill<<<(ND + TB - 1) / TB, blk, 0, stream>>>(bufB, 0.0f, ND);   // after GEMM consumed bufB
    {
        long long waves = (long long)E * 32;
        k_agg<<<(unsigned)((waves + TB - 1) / TB), blk, 0, stream>>>(ei, ew, dinv, bufA, bufB, E);
    }
    k_post2_pool<<<(ND + TB - 1) / TB, blk, 0, stream>>>(bufB, bufA, dinv, b2, batch, pooled, ND);

    // 4) readout
    k_final<<<(NGRAPH * DOUT + TB - 1) / TB, blk, 0, stream>>>(pooled, lw, lb, out);
}